// DpclObjfComputer_51307679318166
// MI455X (gfx1250) — compile-verified
//
#include <hip/hip_runtime.h>

typedef __attribute__((ext_vector_type(2))) float v2f;
typedef __attribute__((ext_vector_type(8))) float v8f;

#define N_UTT 16
#define FT    154200      // 257 * 600
#define DIM   40
#define TLEN  600.0f
#define STEPS 75          // k-steps (of 4) per segment
#define SEGS  514         // segments per utterance; 514*75*4 = 154200
#define GT    1536        // 6 tiles * 256 floats (Gram upper triangle)
#define MT    96          // padded M': 48 d-slots * 2 speakers
#define WSN   (GT + MT)   // floats of workspace per utterance

// ---------------------------------------------------------------------------
// Main pass: per wave, accumulate 6 symmetric 16x16 Gram tiles of
// G' = E^T diag(mix) E with V_WMMA_F32_16X16X4_F32, plus per-lane M' partials.
// Branchless: tile-2 lanes with d>=8 read a safe duplicate address and mask
// the value to exact 0, so the load clause never touches EXEC.
// ---------------------------------------------------------------------------
__global__ __launch_bounds__(32)
void dpcl_gram_kernel(const float* __restrict__ emb,
                      const float* __restrict__ mref,
                      const float* __restrict__ mix,
                      float* __restrict__ ws)
{
    const int n    = blockIdx.y;
    const int seg  = blockIdx.x;
    const int lane = threadIdx.x;     // 0..31 (wave32)
    const int d    = lane & 15;       // column within a 16-wide tile
    const int half = lane >> 4;       // K-slot group (K = v + 2*half)

    const float* pe = emb  + (size_t)n * FT * DIM;
    const float* pm = mix  + (size_t)n * FT;
    const float* pr = mref + (size_t)n * FT * 2;

    const int   kfirst = seg * STEPS * 4 + 2 * half;  // this half's first K row
    const bool  t2ok   = (d < 8);      // tile 2 covers d=32..47; valid while <40
    const float mask2  = t2ok ? 1.0f : 0.0f;

    const float* bp  = pe + (size_t)kfirst * DIM + d;  // tiles 0/1 base
    const float* bp2 = t2ok ? (bp + 32) : bp;          // tile 2 (safe dup addr)
    const float* pmk = pm + kfirst;
    const float* prk = pr + 2 * (size_t)kfirst;

    v8f acc00 = {}, acc01 = {}, acc02 = {}, acc11 = {}, acc12 = {}, acc22 = {};
    float m00 = 0.f, m01 = 0.f, m10 = 0.f, m11 = 0.f, m20 = 0.f, m21 = 0.f;

    for (int s = 0; s < STEPS; ++s) {
        // Embedding fragments: 6 unconditional, coalesced, non-temporal loads
        float a00  = __builtin_nontemporal_load(bp);
        float a01  = __builtin_nontemporal_load(bp + 16);
        float a02r = __builtin_nontemporal_load(bp2);
        float a10  = __builtin_nontemporal_load(bp + DIM);
        float a11  = __builtin_nontemporal_load(bp + DIM + 16);
        float a12r = __builtin_nontemporal_load(bp2 + DIM);
        float a02 = a02r * mask2;
        float a12 = a12r * mask2;

        // Per-K mix weights and speaker classes (argmax over S=2, ties -> 0)
        float w0  = pmk[0], w1 = pmk[1];
        float r00 = prk[0], r01 = prk[1];
        float r10 = prk[2], r11 = prk[3];
        float w0s1 = (r01 > r00) ? w0 : 0.0f;  float w0s0 = w0 - w0s1;
        float w1s1 = (r11 > r10) ? w1 : 0.0f;  float w1s0 = w1 - w1s1;

        // M' partials: this lane owns columns d, d+16, d+32
        m00 += w0s0 * a00 + w1s0 * a10;   m01 += w0s1 * a00 + w1s1 * a10;
        m10 += w0s0 * a01 + w1s0 * a11;   m11 += w0s1 * a01 + w1s1 * a11;
        m20 += w0s0 * a02 + w1s0 * a12;   m21 += w0s1 * a02 + w1s1 * a12;

        // A fragments (raw), B fragments (mix-scaled); identical lane layout:
        // component v holds K = v + 2*half for column/row index d.
        v2f fa0 = { a00, a10 };
        v2f fa1 = { a01, a11 };
        v2f fa2 = { a02, a12 };
        v2f fb0 = { a00 * w0, a10 * w1 };
        v2f fb1 = { a01 * w0, a11 * w1 };
        v2f fb2 = { a02 * w0, a12 * w1 };

        acc00 = __builtin_amdgcn_wmma_f32_16x16x4_f32(false, fa0, false, fb0, (short)0, acc00, false, false);
        acc01 = __builtin_amdgcn_wmma_f32_16x16x4_f32(false, fa0, false, fb1, (short)0, acc01, false, false);
        acc02 = __builtin_amdgcn_wmma_f32_16x16x4_f32(false, fa0, false, fb2, (short)0, acc02, false, false);
        acc11 = __builtin_amdgcn_wmma_f32_16x16x4_f32(false, fa1, false, fb1, (short)0, acc11, false, false);
        acc12 = __builtin_amdgcn_wmma_f32_16x16x4_f32(false, fa1, false, fb2, (short)0, acc12, false, false);
        acc22 = __builtin_amdgcn_wmma_f32_16x16x4_f32(false, fa2, false, fb2, (short)0, acc22, false, false);

        bp  += 4 * DIM;
        bp2 += 4 * DIM;
        pmk += 4;
        prk += 8;
    }

    // Flush: C/D layout is row = r + 8*half, col = d.
    float* gw = ws + (size_t)n * WSN;
#define FLUSH_ACC(P, A)                                                        \
    _Pragma("unroll")                                                          \
    for (int r = 0; r < 8; ++r)                                                \
        atomicAdd(&gw[(P) * 256 + (r + 8 * half) * 16 + d], (A)[r]);
    FLUSH_ACC(0, acc00); FLUSH_ACC(1, acc01); FLUSH_ACC(2, acc02);
    FLUSH_ACC(3, acc11); FLUSH_ACC(4, acc12); FLUSH_ACC(5, acc22);
#undef FLUSH_ACC

    float* mw = gw + GT;  // padded entries (d>=40) accumulate exact zeros
    atomicAdd(&mw[(0 * 16 + d) * 2 + 0], m00);
    atomicAdd(&mw[(0 * 16 + d) * 2 + 1], m01);
    atomicAdd(&mw[(1 * 16 + d) * 2 + 0], m10);
    atomicAdd(&mw[(1 * 16 + d) * 2 + 1], m11);
    atomicAdd(&mw[(2 * 16 + d) * 2 + 0], m20);
    atomicAdd(&mw[(2 * 16 + d) * 2 + 1], m21);
}

// ---------------------------------------------------------------------------
// Finish pass: per utterance compute tot, r_s; square/combine workspace.
// loss_n = (||G'||^2 + sum r'^2 - 2||M'||^2) / tot^2 / T ; mean over N.
// ---------------------------------------------------------------------------
__device__ __forceinline__ float block_reduce(float v, float* sbuf, int tid) {
    sbuf[tid] = v;
    __syncthreads();
    for (int off = 128; off > 0; off >>= 1) {
        if (tid < off) sbuf[tid] += sbuf[tid + off];
        __syncthreads();
    }
    float r = sbuf[0];
    __syncthreads();
    return r;
}

__global__ __launch_bounds__(256)
void dpcl_finish_kernel(const float* __restrict__ mref,
                        const float* __restrict__ mix,
                        const float* __restrict__ ws,
                        float* __restrict__ out)
{
    __shared__ float sbuf[256];
    const int n = blockIdx.x, tid = threadIdx.x;
    const float* pm = mix + (size_t)n * FT;
    const float* pr = mref + (size_t)n * FT * 2;

    float tot = 0.f, r0 = 0.f, r1 = 0.f;
    for (int k = tid; k < FT; k += 256) {
        float w = pm[k];
        tot += w;
        if (pr[2 * k + 1] > pr[2 * k]) r1 += w; else r0 += w;
    }

    const float* gw = ws + (size_t)n * WSN;
    float comb = 0.f;   // gsum - 2*msum partial
    for (int i = tid; i < GT; i += 256) {
        int p = i >> 8;                                  // tile-pair index
        float f = (p == 0 || p == 3 || p == 5) ? 1.0f : 2.0f;  // symmetry
        float g = gw[i];
        comb += f * g * g;
    }
    for (int i = tid; i < MT; i += 256) {
        float m = gw[GT + i];
        comb -= 2.0f * m * m;
    }

    tot  = block_reduce(tot,  sbuf, tid);
    r0   = block_reduce(r0,   sbuf, tid);
    r1   = block_reduce(r1,   sbuf, tid);
    comb = block_reduce(comb, sbuf, tid);

    if (tid == 0) {
        float loss_n = (comb + r0 * r0 + r1 * r1) / (tot * tot) / TLEN;
        atomicAdd(out, loss_n * (1.0f / (float)N_UTT));
    }
}

extern "C" void kernel_launch(void* const* d_in, const int* in_sizes, int n_in,
                              void* d_out, int out_size, void* d_ws, size_t ws_size,
                              hipStream_t stream) {
    const float* emb  = (const float*)d_in[0];  // N x FT x D
    const float* mref = (const float*)d_in[1];  // N x F x T x S
    const float* mix  = (const float*)d_in[2];  // N x F x T
    float* out = (float*)d_out;
    float* ws  = (float*)d_ws;

    hipMemsetAsync(ws, 0, (size_t)N_UTT * WSN * sizeof(float), stream);
    hipMemsetAsync(out, 0, sizeof(float), stream);

    dim3 grid(SEGS, N_UTT);
    dpcl_gram_kernel<<<grid, 32, 0, stream>>>(emb, mref, mix, ws);
    dpcl_finish_kernel<<<N_UTT, 256, 0, stream>>>(mref, mix, ws, out);
}